// simLayer_69922067579426
// MI455X (gfx1250) — compile-verified
//
#include <hip/hip_runtime.h>
#include <hip/hip_bf16.h>
#include <stddef.h>

// ---------------------------------------------------------------------------
// Problem constants (from the reference)
// ---------------------------------------------------------------------------
#define BB_   8
#define HH_   126
#define WW_   126
#define FF_   64
#define DD_   32
#define HP_   128          // H+2
#define WP_   128          // W+2
#define ITERS_ 3
#define NIT_   20

// Float constants derived in double, truncated to float (reference uses f32 arrays).
constexpr double dDT  = 0.1;
constexpr double dRHO = 1.0;
constexpr double dNU  = 0.1;
constexpr double dFRC = 1.0;
constexpr double dDX  = 2.0 / (BB_ + 1);
constexpr double dDY  = 2.0 / (HH_ + 1);
constexpr double dDX2 = dDX * dDX;
constexpr double dDY2 = dDY * dDY;
constexpr double dDEN = 2.0 * (dDX2 + dDY2);

constexpr float cDX2     = (float)dDX2;
constexpr float cDY2     = (float)dDY2;
constexpr float cINVDEN  = (float)(1.0 / dDEN);
constexpr float cCCOEF   = (float)(dDX2 * dDY2 / dDEN);
constexpr float cINV2DX  = (float)(1.0 / (2.0 * dDX));
constexpr float cINV2DY  = (float)(1.0 / (2.0 * dDY));
constexpr float cINVDT   = (float)(1.0 / dDT);
constexpr float cDTDX    = (float)(dDT / dDX);
constexpr float cDTDY    = (float)(dDT / dDY);
constexpr float cPCX     = (float)(dDT / (2.0 * dRHO * dDX));
constexpr float cPCY     = (float)(dDT / (2.0 * dRHO * dDY));
constexpr float cNUDTDX2 = (float)(dNU * dDT / dDX2);
constexpr float cNUDTDY2 = (float)(dNU * dDT / dDY2);
constexpr float cFRCDT   = (float)(dFRC * dDT);
constexpr float cRHO     = (float)dRHO;

typedef float v2f __attribute__((ext_vector_type(2)));
typedef float v8f __attribute__((ext_vector_type(8)));

// ---------------------------------------------------------------------------
// Kernel 1: p_init = pad(x) @ w^T  via V_WMMA_F32_16X16X4_F32.
// GEMM over the *interior* points only (x is contiguous M=8*126*126, K=64).
// Result stored to d_ws as (M, 32) row-major; bias b_lin added later at LDS
// load time (padding ring of p is exactly b_lin).
//
// Lane layouts (ISA 7.12.2, wave32):
//   A 16x4 f32 : m = lane%16 ; VGPR0 holds K = (lane<16 ? 0 : 2), VGPR1 = K+1
//   B  4x16 f32: n = lane%16 ; VGPR0 holds K = (lane<16 ? 0 : 2), VGPR1 = K+1
//   C 16x16 f32: element i -> M = i + (lane<16 ? 0 : 8), N = lane%16
// ---------------------------------------------------------------------------
#define GEMM_M (BB_ * HH_ * WW_)   // 127008 = 16 * 7938 exactly

__global__ __launch_bounds__(32)
void pinit_wmma_kernel(const float* __restrict__ x,   // (M, 64)
                       const float* __restrict__ w,   // (32, 64)
                       float* __restrict__ pws)       // (M, 32)
{
    const int tile = blockIdx.x;            // 7938 tiles of 16 rows
    const int base = tile * 16;
    const int lane = threadIdx.x;           // wave32
    const int m    = lane & 15;
    const int n    = lane & 15;
    const int koff = (lane < 16) ? 0 : 2;

    v8f acc0 = {0.f, 0.f, 0.f, 0.f, 0.f, 0.f, 0.f, 0.f};
    v8f acc1 = {0.f, 0.f, 0.f, 0.f, 0.f, 0.f, 0.f, 0.f};

    const float* arow = x + (size_t)(base + m) * FF_ + koff;
    const float* br0  = w + (size_t)(n)      * FF_ + koff;   // B[k][n] = w[n][f]
    const float* br1  = w + (size_t)(n + 16) * FF_ + koff;

#pragma unroll
    for (int kk = 0; kk < 16; ++kk) {       // K = 64 in steps of 4
        const int f0 = kk * 4;
        v2f a;  a.x  = arow[f0]; a.y  = arow[f0 + 1];
        v2f b0; b0.x = br0[f0];  b0.y = br0[f0 + 1];
        v2f b1; b1.x = br1[f0];  b1.y = br1[f0 + 1];
        acc0 = __builtin_amdgcn_wmma_f32_16x16x4_f32(false, a, false, b0,
                                                     (short)0, acc0, false, false);
        acc1 = __builtin_amdgcn_wmma_f32_16x16x4_f32(false, a, false, b1,
                                                     (short)0, acc1, false, false);
    }

    const int mbase = (lane < 16) ? 0 : 8;
#pragma unroll
    for (int i = 0; i < 8; ++i) {
        const size_t row = (size_t)(base + mbase + i);
        pws[row * DD_ + n]      = acc0[i];
        pws[row * DD_ + n + 16] = acc1[i];
    }
}

// ---------------------------------------------------------------------------
// Kernel 2: full LDS-resident simulation. One workgroup per (b,d) plane.
// U, V, P planes (128x128 f32 = 64 KB each) live in LDS (192 KB total,
// within the 320 KB/workgroup CDNA5 limit). The Jacobi source term `b`
// is thread-private (16 registers). 1024 threads = 32 wave32 waves.
// Thread t owns column c = t%128 and rows r = t/128 + 8k, k = 0..15.
// Columns are periodic (matches reference wrap); rows: p Neumann copy,
// u/v zero at rows 0 and 127.
// ---------------------------------------------------------------------------
__global__ __launch_bounds__(1024)
void sim_lds_kernel(const float* __restrict__ u0,    // (8,128,128,32)
                    const float* __restrict__ v0,    // (8,128,128,32)
                    const float* __restrict__ blin_g,// (32,)
                    const float* __restrict__ pws,   // (8*126*126, 32)
                    float* __restrict__ out)         // (8,128,32)
{
    __shared__ float U[HP_][WP_];
    __shared__ float V[HP_][WP_];
    __shared__ float P[HP_][WP_];

    const int blk = blockIdx.x;            // 256 = 8 * 32
    const int b   = blk >> 5;
    const int d   = blk & 31;

    const int t  = threadIdx.x;            // 0..1023
    const int c  = t & 127;
    const int rg = t >> 7;                 // 0..7
    const int cm = (c == 0)   ? 127 : c - 1;
    const int cp = (c == 127) ? 0   : c + 1;

    const float blin = blin_g[d];

    // ---- load planes into LDS ----
#pragma unroll
    for (int k = 0; k < 16; ++k) {
        const int r = rg + 8 * k;
        const size_t g = ((((size_t)b * HP_ + r) * WP_) + c) * DD_ + d;
        U[r][c] = u0[g];
        V[r][c] = v0[g];
        float pv = blin;
        if (r >= 1 && r <= HH_ && c >= 1 && c <= WW_) {
            pv += pws[((((size_t)b * HH_ + (r - 1)) * WW_) + (c - 1)) * DD_ + d];
        }
        P[r][c] = pv;
    }
    __syncthreads();

    for (int it = 0; it < ITERS_; ++it) {
        // ---- build_b into registers (rows 1..126; rows 0,127 are zero) ----
        float bb[16];
#pragma unroll
        for (int k = 0; k < 16; ++k) {
            const int r = rg + 8 * k;
            if (r >= 1 && r <= HH_) {
                const float du_dx = (U[r][cp]     - U[r][cm])     * cINV2DX;
                const float dv_dy = (V[r + 1][c]  - V[r - 1][c])  * cINV2DY;
                const float du_dy = (U[r + 1][c]  - U[r - 1][c])  * cINV2DY;
                const float dv_dx = (V[r][cp]     - V[r][cm])     * cINV2DX;
                bb[k] = cRHO * ((du_dx + dv_dy) * cINVDT
                                - du_dx * du_dx
                                - 2.0f * du_dy * dv_dx
                                - dv_dy * dv_dy);
            } else {
                bb[k] = 0.0f;
            }
        }

        // ---- pressure Poisson: 20 Jacobi sweeps, barrier-split in-place ----
        for (int jt = 0; jt < NIT_; ++jt) {
            float pn[16];
#pragma unroll
            for (int k = 0; k < 16; ++k) {
                const int r = rg + 8 * k;
                if (r >= 1 && r <= HH_) {
                    pn[k] = ((P[r][cp] + P[r][cm]) * cDY2 +
                             (P[r + 1][c] + P[r - 1][c]) * cDX2) * cINVDEN
                            - cCCOEF * bb[k];
                } else {
                    pn[k] = 0.0f;
                }
            }
            __syncthreads();
#pragma unroll
            for (int k = 0; k < 16; ++k) {
                const int r = rg + 8 * k;
                if (r >= 1 && r <= HH_) P[r][c] = pn[k];
            }
            __syncthreads();
            // Neumann rows, using the freshly updated interior
            if (rg == 0) P[0][c]   = P[1][c];
            if (rg == 7) P[127][c] = P[126][c];
            __syncthreads();
        }

        // ---- velocity step, register-staged (reads old U,V,P; writes U,V) ----
        float un[16], vn[16];
#pragma unroll
        for (int k = 0; k < 16; ++k) {
            const int r = rg + 8 * k;
            if (r >= 1 && r <= HH_) {
                const float uc = U[r][c];
                const float vc = V[r][c];
                const float uxm = U[r][cm],    uxp = U[r][cp];
                const float uym = U[r - 1][c], uyp = U[r + 1][c];
                const float vxm = V[r][cm],    vxp = V[r][cp];
                const float vym = V[r - 1][c], vyp = V[r + 1][c];
                un[k] = uc
                        - uc * cDTDX * (uc - uxm)
                        - vc * cDTDY * (uc - uym)
                        - cPCX * (P[r][cp] - P[r][cm])
                        + cNUDTDX2 * (uxp - 2.0f * uc + uxm)
                        + cNUDTDY2 * (uyp - 2.0f * uc + uym)
                        + cFRCDT;
                vn[k] = vc
                        - uc * cDTDX * (vc - vxm)
                        - vc * cDTDY * (vc - vym)
                        - cPCY * (P[r + 1][c] - P[r - 1][c])
                        + cNUDTDX2 * (vxp - 2.0f * vc + vxm)
                        + cNUDTDY2 * (vyp - 2.0f * vc + vym);
            } else {
                un[k] = 0.0f;   // u,v rows 0 and 127 forced to zero
                vn[k] = 0.0f;
            }
        }
        __syncthreads();
#pragma unroll
        for (int k = 0; k < 16; ++k) {
            const int r = rg + 8 * k;
            U[r][c] = un[k];
            V[r][c] = vn[k];
        }
        __syncthreads();
    }

    // ---- output: p[:, -1, :, :]  -> (8, 128, 32) ----
    if (t < WP_) {
        out[(((size_t)b * WP_) + t) * DD_ + d] = P[127][t];
    }
}

// ---------------------------------------------------------------------------
// Launch: inputs in setup_inputs() order: x, u0, v0, w, b_lin (all f32).
// d_ws needs GEMM_M * 32 * 4 = ~16.3 MB for the p_init staging buffer.
// ---------------------------------------------------------------------------
extern "C" void kernel_launch(void* const* d_in, const int* in_sizes, int n_in,
                              void* d_out, int out_size, void* d_ws, size_t ws_size,
                              hipStream_t stream) {
    (void)in_sizes; (void)n_in; (void)out_size; (void)ws_size;
    const float* x    = (const float*)d_in[0];
    const float* u0   = (const float*)d_in[1];
    const float* v0   = (const float*)d_in[2];
    const float* w    = (const float*)d_in[3];
    const float* blin = (const float*)d_in[4];
    float* out = (float*)d_out;
    float* pws = (float*)d_ws;

    // 1) WMMA fp32 GEMM: p_init interior = x @ w^T  (bias folded in later)
    pinit_wmma_kernel<<<GEMM_M / 16, 32, 0, stream>>>(x, w, pws);

    // 2) LDS-resident solver: one workgroup per (b,d) plane
    sim_lds_kernel<<<BB_ * DD_, 1024, 0, stream>>>(u0, v0, blin, pws, out);
}